// SpatialAttention_37142877175987
// MI455X (gfx1250) — compile-verified
//
#include <hip/hip_runtime.h>
#include <hip/hip_bf16.h>

typedef float v2f __attribute__((ext_vector_type(2)));
typedef float v4f __attribute__((ext_vector_type(4)));
typedef float v8f __attribute__((ext_vector_type(8)));

#define CCH   256
#define HW    56
#define IMGPX (HW * HW)   // 3136 pixels per image

// ---------------------------------------------------------------------------
// Kernel 1: channel-wise avg+max pooling.
// One wave32 per pixel: each lane loads 2x float4 (b128) fully coalesced
// (1 KB per pixel), reduces 8 values in-lane, then a 5-step __shfl_xor tree.
// x stays resident in the 192MB L2 for the scale pass.
// ---------------------------------------------------------------------------
__global__ void __launch_bounds__(256)
pool_kernel(const float* __restrict__ x, float2* __restrict__ pooled, int npix) {
  const int lane = threadIdx.x & 31;
  const int wave = (int)((blockIdx.x * blockDim.x + threadIdx.x) >> 5);
  if (wave >= npix) return;  // wave-uniform

  const float4* xp = (const float4*)(x + (size_t)wave * CCH);
  // look-ahead prefetch (lowers to global_prefetch_b8)
  if (wave + 64 < npix)
    __builtin_prefetch(x + (size_t)(wave + 64) * CCH + lane * 8, 0, 1);

  float4 a = xp[lane];        // floats [0,128)
  float4 b = xp[lane + 32];   // floats [128,256)

  float s = ((a.x + a.y) + (a.z + a.w)) + ((b.x + b.y) + (b.z + b.w));
  float m = fmaxf(fmaxf(fmaxf(a.x, a.y), fmaxf(a.z, a.w)),
                  fmaxf(fmaxf(b.x, b.y), fmaxf(b.z, b.w)));
#pragma unroll
  for (int off = 16; off >= 1; off >>= 1) {
    s += __shfl_xor(s, off, 32);
    m = fmaxf(m, __shfl_xor(m, off, 32));
  }
  if (lane == 0) pooled[wave] = make_float2(s * (1.0f / 256.0f), m);
}

// ---------------------------------------------------------------------------
// Kernel 2: 7x7 conv (2->1 ch, SAME) + sigmoid as im2col GEMM on
// V_WMMA_F32_16X16X4_F32. One wave computes 16 output pixels.
// K = 98 taps (49 spatial positions x 2 channels) padded to 100 -> 25 WMMA
// steps of K=4. Key structure: a lane's two A elements (j0, j0+1) share one
// spatial tap s = k0/2 + half with ch = 0,1  =>  one float2 gather from the
// pooled map gives A directly, one float2 load gives the B weight pair.
// Fully unrolled so all dy/dx decompositions constant-fold (no divisions).
// EXEC stays all-1s across all WMMA issues (masking via selects only).
// ---------------------------------------------------------------------------
__global__ void __launch_bounds__(256)
conv_wmma_kernel(const float2* __restrict__ pooled,
                 const float* __restrict__ wflat,   // 7*7*2 = 98 floats
                 float* __restrict__ attn, int npix) {
  const int lane   = threadIdx.x & 31;
  const int waveId = (int)((blockIdx.x * blockDim.x + threadIdx.x) >> 5);
  const int pixBase = waveId * 16;
  if (pixBase >= npix) return;  // wave-uniform

  const int half = lane >> 4;          // 0: K rows {k0,k0+1}; 1: {k0+2,k0+3}
  const int mrow = lane & 15;          // pixel row of A held by this lane
  int p = pixBase + mrow;
  p = p < npix ? p : npix - 1;
  const int b_   = p / IMGPX;
  const int rem  = p - b_ * IMGPX;
  const int h    = rem / HW;
  const int w    = rem - h * HW;
  const int ibase = b_ * IMGPX;

  const float2* wpair = (const float2*)wflat;  // 49 (dy,dx) -> {w_ch0, w_ch1}

  v8f acc = {0.f, 0.f, 0.f, 0.f, 0.f, 0.f, 0.f, 0.f};
#pragma unroll
  for (int k0 = 0; k0 < 100; k0 += 4) {
    const int s0 = k0 >> 1;            // spatial tap for half==0 (<= 48)
    const int s1 = s0 + 1;             // spatial tap for half==1 (<= 49)
    // compile-time after unroll:
    const int dy0 = s0 / 7 - 3, dx0 = s0 % 7 - 3;
    const int dy1 = s1 / 7 - 3, dx1 = s1 % 7 - 3;
    const bool ok1 = (s1 < 49);        // false only for k0 == 96

    const int  dy  = half ? dy1 : dy0;
    const int  dx  = half ? dx1 : dx0;
    const bool sok = half ? ok1 : true;

    const int hh = h + dy;
    const int ww = w + dx;
    const bool ok = sok & ((unsigned)hh < (unsigned)HW) & ((unsigned)ww < (unsigned)HW);

    const int pidx = ok ? (ibase + hh * HW + ww) : 0;
    const float2 pv = pooled[pidx];
    v2f A = { ok ? pv.x : 0.0f, ok ? pv.y : 0.0f };

    const int sIdx = sok ? (s0 + half) : 0;
    const float2 wv = wpair[sIdx];
    v2f Bv = { sok ? wv.x : 0.0f, sok ? wv.y : 0.0f };

    acc = __builtin_amdgcn_wmma_f32_16x16x4_f32(
        /*neg_a=*/false, A, /*neg_b=*/false, Bv,
        /*c_mod=*/(short)0, acc, /*reuse_a=*/false, /*reuse_b=*/false);
  }

  // All 16 D columns are identical; lane 0 holds rows 0..7 (acc[r] = D[r][0]),
  // lane 16 holds rows 8..15.
  if ((lane & 15) == 0) {
    const int base = pixBase + half * 8;
#pragma unroll
    for (int r = 0; r < 8; ++r) {
      const int pp = base + r;
      if (pp < npix) attn[pp] = 1.0f / (1.0f + __expf(-acc[r]));
    }
  }
}

// ---------------------------------------------------------------------------
// Kernel 3: out = x * attn (broadcast over 256 channels).
// Pure stream: b128 loads of x (should hit L2 after kernel 1),
// non-temporal b128 stores of out (never re-read; don't evict x).
// ---------------------------------------------------------------------------
__global__ void __launch_bounds__(256)
scale_kernel(const float* __restrict__ x, const float* __restrict__ attn,
             float* __restrict__ out, int n4) {
  const int idx = (int)(blockIdx.x * blockDim.x + threadIdx.x);
  if (idx >= n4) return;
  const v4f* x4 = (const v4f*)x;
  v4f* o4 = (v4f*)out;

  const float a = attn[idx >> 6];              // 64 float4s per pixel
  v4f v = __builtin_nontemporal_load(&x4[idx]);
  v4f r = v * a;
  __builtin_nontemporal_store(r, &o4[idx]);
}

// ---------------------------------------------------------------------------
extern "C" void kernel_launch(void* const* d_in, const int* in_sizes, int n_in,
                              void* d_out, int out_size, void* d_ws, size_t ws_size,
                              hipStream_t stream) {
  (void)n_in; (void)out_size; (void)ws_size;
  const float* x  = (const float*)d_in[0];
  const float* cw = (const float*)d_in[1];          // 98 floats (7,7,2,1)
  float* out = (float*)d_out;

  const int total = in_sizes[0];                    // 25,690,112
  const int npix  = total / CCH;                    // 100,352

  float2* pooled = (float2*)d_ws;                                   // npix * 8 B
  float*  attn   = (float*)((char*)d_ws + (size_t)npix * sizeof(float2)); // npix * 4 B

  {  // pool: one wave per pixel
    const long long threads = (long long)npix * 32;
    const int blocks = (int)((threads + 255) / 256);
    pool_kernel<<<blocks, 256, 0, stream>>>(x, pooled, npix);
  }
  {  // conv+sigmoid: one wave per 16 pixels
    const int waves = (npix + 15) / 16;
    const long long threads = (long long)waves * 32;
    const int blocks = (int)((threads + 255) / 256);
    conv_wmma_kernel<<<blocks, 256, 0, stream>>>(pooled, cw, attn, npix);
  }
  {  // scale: one thread per float4
    const int n4 = total / 4;
    const int blocks = (n4 + 255) / 256;
    scale_kernel<<<blocks, 256, 0, stream>>>(x, attn, out, n4);
  }
}